// Module2Network_22608707846768
// MI455X (gfx1250) — compile-verified
//
#include <hip/hip_runtime.h>
#include <math.h>

#define EPS 1e-8f
#define DHEAD 128
#define NFREQ 64
#define NBINS 128

typedef __attribute__((ext_vector_type(2))) float v2f;
typedef __attribute__((ext_vector_type(8))) float v8f;

// Raw hardware v_sqrt_f32 (~1 ulp). Safe here: operands are >= EPS=1e-8
// (never denormal / negative), so the libm denorm-rescale path is dead weight.
__device__ __forceinline__ float fast_sqrtf(float x) {
    return __builtin_amdgcn_sqrtf(x);
}

// ---------------- workspace layout (floats) ----------------
// kfrag: pre-swizzled WMMA B fragments. 48 K-steps (32 dot + 16 mag) x 8 ntiles
//        x 32 lanes x float2  = 24576 floats (96 KB)
// qtab : float4 per (freq,bin): {rp0, rp1, eff_w, q_mag_w} = 32768 floats (128 KB)
#define KFRAG_FLOATS (48 * 8 * 32 * 2)
#define QTAB_F4      (NFREQ * NBINS)

// =====================================================================
// Kernel 0: precompute rotated probes, swizzled B fragments, q-table
// =====================================================================
__global__ void precompute_kernel(const float* __restrict__ probes,
                                  const float* __restrict__ angles,
                                  const float* __restrict__ qwr,
                                  const float* __restrict__ qmw,
                                  const float* __restrict__ kmw,
                                  float* __restrict__ kfrag,
                                  float4* __restrict__ qtab) {
    extern __shared__ float s_rp[];   // 128 x 128 rotated probes (64 KB)
    const int tid = threadIdx.x;

    if (tid < NBINS) {
        const int b = tid;
        float ss = 0.f;
        for (int d = 0; d < DHEAD; ++d) {
            float v = probes[b * DHEAD + d];
            ss += v * v;
        }
        const float inv = 1.0f / (sqrtf(ss) + EPS);
        for (int f = 0; f < NFREQ; ++f) {
            float x = probes[b * DHEAD + 2 * f]     * inv;
            float y = probes[b * DHEAD + 2 * f + 1] * inv;
            float c = cosf(angles[f]);
            float s = sinf(angles[f]);
            float r0 = x * c - y * s;
            float r1 = x * s + y * c;
            s_rp[b * DHEAD + 2 * f]     = r0;
            s_rp[b * DHEAD + 2 * f + 1] = r1;
            // eff_w = -softplus(qwr), numerically stable
            float w  = qwr[b * NFREQ + f];
            float sp = (w > 20.f) ? w : log1pf(expf(w));
            qtab[f * NBINS + b] = make_float4(r0, r1, -sp, qmw[b * NFREQ + f]);
        }
    }
    __syncthreads();

    // Build per-lane-swizzled B fragments for V_WMMA_F32_16X16X4_F32.
    // B (4x16, KxN): VGPR0 = {K=0 lanes0-15, K=2 lanes16-31}, VGPR1 = {K=1, K=3}.
    for (int i = tid; i < 48 * 8 * 32; i += 256) {
        const int step = i >> 8;          // 8 ntiles * 32 lanes = 256 per step
        const int rem  = i & 255;
        const int nt   = rem >> 5;
        const int lane = rem & 31;
        const int b    = nt * 16 + (lane & 15);
        const int hi   = lane >> 4;
        float v0, v1;
        if (step < 32) {                  // dot-product dims: B[d][b] = rp[b][d]
            const int d0 = step * 4 + hi * 2;
            v0 = s_rp[b * DHEAD + d0];
            v1 = s_rp[b * DHEAD + d0 + 1];
        } else {                          // magnitude dims: B[f][b] = kmw[b][f]
            const int f0 = (step - 32) * 4 + hi * 2;
            v0 = kmw[b * NFREQ + f0];
            v1 = kmw[b * NFREQ + f0 + 1];
        }
        kfrag[i * 2]     = v0;
        kfrag[i * 2 + 1] = v1;
    }
}

// =====================================================================
// Kernel 1: k_logits = K @ rp^T + K_mag @ kmw^T + bias  (fused WMMA GEMM)
// 256 threads = 8 waves; each wave owns a 16-row tile of K per iteration.
// =====================================================================
__global__ void klogits_kernel(const float* __restrict__ Kin,
                               const float* __restrict__ kfrag,
                               const float* __restrict__ kbias,
                               float* __restrict__ kout,
                               int numTiles, int itersPerWave, int totalWaves) {
    extern __shared__ float smem[];
    float* s_frag = smem;                        // 24576 floats (96 KB)
    float* s_mag  = smem + KFRAG_FLOATS;         // 8 waves * 16 rows * 64 f (32 KB)
    float* s_bias = s_mag + 8 * 16 * 64;         // 128 floats

    const int tid = threadIdx.x;
    // stage constants into LDS
    for (int i = tid; i < KFRAG_FLOATS / 4; i += 256)
        ((float4*)s_frag)[i] = ((const float4*)kfrag)[i];
    if (tid < NBINS) s_bias[tid] = kbias[tid];
    __syncthreads();

    const int lane = tid & 31;
    const int wave = tid >> 5;
    const int hi   = lane >> 4;
    const int lo   = lane & 15;
    const float2* sfrag2 = (const float2*)s_frag;
    float* magrow = s_mag + wave * 1024 + lo * 64;   // this lane's row (M=lo)

    const int gwid = blockIdx.x * 8 + wave;
    const v8f vzero = {0.f, 0.f, 0.f, 0.f, 0.f, 0.f, 0.f, 0.f};

    for (int it = 0; it < itersPerWave; ++it) {
        const int tile   = gwid + it * totalWaves;
        const bool active = (tile < numTiles);
        const int t0     = active ? tile : 0;

        const float* kp = Kin + (size_t)(t0 * 16 + lo) * DHEAD + hi * 2;

        // Prefetch next iteration's 16x128 f32 A-tile (8 KB, lane-strided to
        // cover the whole tile in cachelines). Clamped: no divergence.
        {
            int tn = gwid + (it + 1) * totalWaves;
            if (tn >= numTiles) tn = numTiles - 1;
            const char* np = (const char*)(Kin + (size_t)tn * 16 * DHEAD);
            __builtin_prefetch(np + lane * 256, 0, 3);
        }

        v8f acc[8];
#pragma unroll
        for (int n = 0; n < 8; ++n) acc[n] = vzero;

        // --- phase 1: dot-product dims (K=0..127), 32 WMMA steps of K=4 ---
#pragma unroll 4
        for (int t = 0; t < 32; ++t) {
            float2 a = *(const float2*)(kp + t * 4);   // A: rows lo, dims t*4+hi*2..+1
            // per-pair magnitude for the fused mag-GEMM (freq = 2t+hi of row lo)
            magrow[2 * t + hi] = fast_sqrtf(a.x * a.x + a.y * a.y + EPS);
            v2f av = {a.x, a.y};
            const float2* bp = sfrag2 + t * 256 + lane;
#pragma unroll
            for (int nt = 0; nt < 8; ++nt) {
                float2 bf = bp[nt * 32];
                v2f bv = {bf.x, bf.y};
                acc[nt] = __builtin_amdgcn_wmma_f32_16x16x4_f32(
                    false, av, false, bv, (short)0, acc[nt], false, false);
            }
        }
        __syncthreads();   // cross-lane magnitude visibility

        // --- phase 2: magnitude dims (F=0..63), 16 WMMA steps of K=4 ---
#pragma unroll 2
        for (int s = 0; s < 16; ++s) {
            float2 m = *(const float2*)(magrow + s * 4 + hi * 2);
            v2f av = {m.x, m.y};
            const float2* bp = sfrag2 + (32 + s) * 256 + lane;
#pragma unroll
            for (int nt = 0; nt < 8; ++nt) {
                float2 bf = bp[nt * 32];
                v2f bv = {bf.x, bf.y};
                acc[nt] = __builtin_amdgcn_wmma_f32_16x16x4_f32(
                    false, av, false, bv, (short)0, acc[nt], false, false);
            }
        }

        // --- epilogue: + bias, store. C/D layout: VGPR v -> M = v + hi*8 ---
        if (active) {
#pragma unroll
            for (int nt = 0; nt < 8; ++nt) {
                const float bias = s_bias[nt * 16 + lo];
#pragma unroll
                for (int v = 0; v < 8; ++v) {
                    const size_t r = (size_t)tile * 16 + hi * 8 + v;
                    kout[r * NBINS + nt * 16 + lo] = acc[nt][v] + bias;
                }
            }
        }
        __syncthreads();   // protect s_mag before next iteration rewrites it
    }
}

// =====================================================================
// Kernel 2: q_logits (distance scoring, sqrt inside reduction -> TRANS)
// 256 threads: 2 queries x 128 bins per iteration; LDS-resident table.
// =====================================================================
__global__ void qlogits_kernel(const float* __restrict__ Q,
                               const float4* __restrict__ qtab,
                               const float* __restrict__ qbias,
                               float* __restrict__ qout,
                               int numQ, int itersPerBlock, int numBlocksTot) {
    extern __shared__ float smem[];
    float4* s_tab  = (float4*)smem;              // 8192 float4 (128 KB)
    float*  s_qn   = smem + 4 * QTAB_F4;         // 2 x 128 normalized Q
    float*  s_qm   = s_qn + 2 * DHEAD;           // 2 x 64 Q magnitudes
    float*  s_red  = s_qm + 2 * NFREQ;           // 8 wave partials
    float*  s_ninv = s_red + 8;                  // 2 inverse norms

    const int tid  = threadIdx.x;
    const int lane = tid & 31;
    const int wave = tid >> 5;

    for (int i = tid; i < QTAB_F4; i += 256) s_tab[i] = qtab[i];
    __syncthreads();

    for (int it = 0; it < itersPerBlock; ++it) {
        const int q0 = (blockIdx.x + it * numBlocksTot) * 2;
        const bool active = (q0 + 1 < numQ);
        const int qq = active ? q0 : 0;

        // ---- phase 1: L2-normalize the 2 queries, per-freq magnitudes ----
        {
            const int half = tid >> 7;
            const int d    = tid & 127;
            const float v  = Q[(size_t)(qq + half) * DHEAD + d];
            float sq = v * v;
            for (int off = 16; off > 0; off >>= 1) sq += __shfl_xor(sq, off, 32);
            if (lane == 0) s_red[wave] = sq;
            __syncthreads();
            if (tid < 2) {
                float s = s_red[tid * 4] + s_red[tid * 4 + 1] +
                          s_red[tid * 4 + 2] + s_red[tid * 4 + 3];
                s_ninv[tid] = 1.0f / (sqrtf(s) + EPS);
            }
            __syncthreads();
            s_qn[half * DHEAD + d] = v * s_ninv[half];
            __syncthreads();
            if (tid < 128) {
                const int h2 = tid >> 6, f = tid & 63;
                float qx = s_qn[h2 * DHEAD + 2 * f];
                float qy = s_qn[h2 * DHEAD + 2 * f + 1];
                s_qm[h2 * NFREQ + f] = fast_sqrtf(qx * qx + qy * qy + EPS);
            }
            __syncthreads();
        }

        // ---- phase 2: one lane per (query, bin), sequential freq loop ----
        {
            const int half = tid >> 7;
            const int bin  = tid & 127;
            const float* qn = s_qn + half * DHEAD;
            const float* qm = s_qm + half * NFREQ;
            float acc = 0.f;
#pragma unroll 4
            for (int f = 0; f < NFREQ; ++f) {
                float4 tb = s_tab[f * NBINS + bin];
                float dx = tb.x - qn[2 * f];
                float dy = tb.y - qn[2 * f + 1];
                acc += fast_sqrtf(dx * dx + dy * dy + EPS) * tb.z + qm[f] * tb.w;
            }
            if (active)
                qout[(size_t)(qq + half) * NBINS + bin] = acc + qbias[bin];
        }
        __syncthreads();   // protect s_qn/s_qm for next iteration
    }
}

// =====================================================================
extern "C" void kernel_launch(void* const* d_in, const int* in_sizes, int n_in,
                              void* d_out, int out_size, void* d_ws, size_t ws_size,
                              hipStream_t stream) {
    const float* Q      = (const float*)d_in[0];
    const float* K      = (const float*)d_in[1];
    const float* angles = (const float*)d_in[2];
    const float* probes = (const float*)d_in[3];
    const float* kmw    = (const float*)d_in[4];
    const float* kbias  = (const float*)d_in[5];
    const float* qwr    = (const float*)d_in[6];
    const float* qmw    = (const float*)d_in[7];
    const float* qmbias = (const float*)d_in[8];

    const int NQ = in_sizes[0] / DHEAD;   // 8192
    const int NK = in_sizes[1] / DHEAD;   // 524288

    float*  ws    = (float*)d_ws;
    float*  kfrag = ws;                               // 96 KB
    float4* qtab  = (float4*)(ws + KFRAG_FLOATS);     // 128 KB (16B aligned)

    float* kout = (float*)d_out;
    float* qout = kout + (size_t)NK * NBINS;

    // 0) precompute constants
    precompute_kernel<<<1, 256, DHEAD * NBINS * sizeof(float), stream>>>(
        probes, angles, qwr, qmw, kmw, kfrag, qtab);

    // 1) k_logits via fused f32 WMMA GEMM
    const int kblocks    = 1024;
    const int totalWaves = kblocks * 8;
    const int numTiles   = NK / 16;
    const int itersK     = (numTiles + totalWaves - 1) / totalWaves;
    const size_t kshmem  = (KFRAG_FLOATS + 8 * 16 * 64 + NBINS) * sizeof(float);
    klogits_kernel<<<kblocks, 256, kshmem, stream>>>(
        K, kfrag, kbias, kout, numTiles, itersK, totalWaves);

    // 2) q_logits (VALU/transcendental)
    const int qblocks   = 256;
    const int pairs     = NQ / 2;
    const int itersQ    = (pairs + qblocks - 1) / qblocks;
    const size_t qshmem = (4 * QTAB_F4 + 2 * DHEAD + 2 * NFREQ + 8 + 2) * sizeof(float);
    qlogits_kernel<<<qblocks, 256, qshmem, stream>>>(
        Q, qtab, qmbias, qout, NQ, itersQ, qblocks);
}